// AttentionSortNet_22265110463114
// MI455X (gfx1250) — compile-verified
//
#include <hip/hip_runtime.h>
#include <hip/hip_bf16.h>

typedef __attribute__((ext_vector_type(2))) float v2f;
typedef __attribute__((ext_vector_type(8))) float v8f;

#define T_LEN 4096
#define NB 64
#define DH 64
#define STR 65   // padded LDS row stride (floats): bank = (row + col) % 64

// LDS layout (floats):
// [0,4160)       q_acc  (64 x STR)
// [4160,8320)    k_acc
// [8320,12480)   r matrix (phases 2-4) / segment ids as int (phase 1)
// [12480,12736)  red   (64 rows x 4 parts)
// [12736,12992)  red2
#define OFF_K   4160
#define OFF_R   8320
#define OFF_RED 12480
#define OFF_RED2 12736
#define SMEM_FLOATS 12992

__launch_bounds__(256)
__global__ void sortnet_fused_kernel(const float* __restrict__ q,
                                     const float* __restrict__ k,
                                     const int* __restrict__ seg,
                                     const float* __restrict__ u,
                                     float* __restrict__ out)
{
    __shared__ float smem[SMEM_FLOATS];
    float* q_acc = smem;
    float* k_acc = smem + OFF_K;
    float* rmat  = smem + OFF_R;
    int*   segl  = reinterpret_cast<int*>(smem + OFF_R);
    float* red   = smem + OFF_RED;
    float* red2  = smem + OFF_RED2;

    const int tid = threadIdx.x;
    const int b   = blockIdx.x;

    // zero bucket accumulators, stage segment ids into LDS
    for (int i = tid; i < OFF_R; i += 256) smem[i] = 0.0f;
    const int* segg = seg + (size_t)b * T_LEN;
    for (int i = tid; i < T_LEN; i += 256) segl[i] = segg[i];
    __syncthreads();

    // ---------- Phase 1: segment sums (streaming, register-accumulated) ----------
    {
        const int lane = tid & 31;     // 32 lanes cover d = 0..63 as float2
        const int c    = tid >> 5;     // 8 chunks of 512 timesteps
        const int d    = lane * 2;
        const int t0 = c * 512, t1 = t0 + 512;
        const float2* q2 = reinterpret_cast<const float2*>(q + (size_t)b * T_LEN * DH);
        const float2* k2 = reinterpret_cast<const float2*>(k + (size_t)b * T_LEN * DH);

        float2 qa = make_float2(0.f, 0.f);
        float2 ka = make_float2(0.f, 0.f);
        int cur = segl[t0];

        auto flush = [&](int s) {
            atomicAdd(&q_acc[s * STR + d],     qa.x);
            atomicAdd(&q_acc[s * STR + d + 1], qa.y);
            atomicAdd(&k_acc[s * STR + d],     ka.x);
            atomicAdd(&k_acc[s * STR + d + 1], ka.y);
            qa = make_float2(0.f, 0.f);
            ka = make_float2(0.f, 0.f);
        };

        for (int t = t0; t < t1; t += 8) {
            float2 qv[8], kv[8];
            int sv[8];
#pragma unroll
            for (int uI = 0; uI < 8; ++uI) {
                qv[uI] = q2[(size_t)(t + uI) * 32 + lane];
                kv[uI] = k2[(size_t)(t + uI) * 32 + lane];
            }
#pragma unroll
            for (int uI = 0; uI < 8; ++uI) sv[uI] = segl[t + uI];
#pragma unroll
            for (int uI = 0; uI < 8; ++uI) {
                if (sv[uI] != cur) { flush(cur); cur = sv[uI]; }
                qa.x += qv[uI].x; qa.y += qv[uI].y;
                ka.x += kv[uI].x; ka.y += kv[uI].y;
            }
        }
        flush(cur);
    }
    __syncthreads();

    // ---------- Phase 2: R = q_sums * k_sums^T via V_WMMA_F32_16X16X4_F32 ----------
    // + fused scale/relu/log/gumbel/temperature epilogue into rmat
    {
        const int lane = tid & 31;
        const int wave = tid >> 5;           // 8 waves, 16 tiles -> 2 tiles/wave
        const int half = lane >> 4;          // K sub-pair select per ISA layout
        const int lm   = lane & 15;
        const float* ug = u + (size_t)b * NB * NB;

        for (int tile = wave; tile < 16; tile += 8) {
            const int mt = (tile >> 2) * 16;
            const int nt = (tile & 3) * 16;
            v8f acc = {0.f, 0.f, 0.f, 0.f, 0.f, 0.f, 0.f, 0.f};
#pragma unroll
            for (int kk = 0; kk < DH; kk += 4) {
                v2f a, bb;
                // A (16x4): lanes 0-15 hold K=kk,kk+1; lanes 16-31 hold K=kk+2,kk+3
                a.x  = q_acc[(mt + lm) * STR + kk + 2 * half];
                a.y  = q_acc[(mt + lm) * STR + kk + 2 * half + 1];
                // B (4x16) = k_sums^T: lane n holds k_sums[n][K]
                bb.x = k_acc[(nt + lm) * STR + kk + 2 * half];
                bb.y = k_acc[(nt + lm) * STR + kk + 2 * half + 1];
                acc = __builtin_amdgcn_wmma_f32_16x16x4_f32(
                        false, a, false, bb, (short)0, acc, false, false);
            }
#pragma unroll
            for (int i = 0; i < 8; ++i) {
                const int row = mt + i + half * 8;
                const int col = nt + lm;
                float rv  = fmaxf(acc[i] * 0.125f, 0.0f);           // * 1/sqrt(64), relu
                float uv  = ug[row * NB + col];
                float gum = -__logf(-__logf(uv + 1e-6f) + 1e-6f);   // gumbel noise
                rmat[row * STR + col] = (__logf(rv + 1e-6f) + gum) * (1.0f / 0.7f);
            }
        }
    }
    __syncthreads();

    // ---------- Phase 3: 8 Sinkhorn iterations (row LSE then column LSE) ----------
    {
        const int idx  = tid >> 2;   // row (or column) index 0..63
        const int part = tid & 3;    // 16-element chunk owner

        for (int it = 0; it < 8; ++it) {
            // ---- rows: subtract logsumexp over columns ----
            {
                const int base = idx * STR + part * 16;
                float m = -1e30f;
#pragma unroll
                for (int j = 0; j < 16; ++j) m = fmaxf(m, rmat[base + j]);
                red[tid] = m;
                __syncthreads();
                float mm = fmaxf(fmaxf(red[idx * 4], red[idx * 4 + 1]),
                                 fmaxf(red[idx * 4 + 2], red[idx * 4 + 3]));
                float s = 0.f;
#pragma unroll
                for (int j = 0; j < 16; ++j) s += __expf(rmat[base + j] - mm);
                red2[tid] = s;
                __syncthreads();
                float tot = red2[idx * 4] + red2[idx * 4 + 1] +
                            red2[idx * 4 + 2] + red2[idx * 4 + 3];
                float lse = mm + __logf(tot);
#pragma unroll
                for (int j = 0; j < 16; ++j) rmat[base + j] -= lse;
                __syncthreads();
            }
            // ---- columns: subtract logsumexp over rows ----
            {
                const int col = idx;
                float m = -1e30f;
#pragma unroll
                for (int j = 0; j < 16; ++j)
                    m = fmaxf(m, rmat[(part * 16 + j) * STR + col]);
                red[tid] = m;
                __syncthreads();
                float mm = fmaxf(fmaxf(red[idx * 4], red[idx * 4 + 1]),
                                 fmaxf(red[idx * 4 + 2], red[idx * 4 + 3]));
                float s = 0.f;
#pragma unroll
                for (int j = 0; j < 16; ++j)
                    s += __expf(rmat[(part * 16 + j) * STR + col] - mm);
                red2[tid] = s;
                __syncthreads();
                float tot = red2[idx * 4] + red2[idx * 4 + 1] +
                            red2[idx * 4 + 2] + red2[idx * 4 + 3];
                float lse = mm + __logf(tot);
#pragma unroll
                for (int j = 0; j < 16; ++j)
                    rmat[(part * 16 + j) * STR + col] -= lse;
                __syncthreads();
            }
        }
    }

    // ---------- Phase 4: out = exp(r), coalesced store ----------
    {
        float* outb = out + (size_t)b * NB * NB;
        for (int e = tid; e < NB * NB; e += 256) {
            const int i = e >> 6, j = e & 63;
            outb[e] = __expf(rmat[i * STR + j]);
        }
    }
}

extern "C" void kernel_launch(void* const* d_in, const int* in_sizes, int n_in,
                              void* d_out, int out_size, void* d_ws, size_t ws_size,
                              hipStream_t stream) {
    const float* q   = (const float*)d_in[0];
    const float* k   = (const float*)d_in[1];
    const int*   seg = (const int*)d_in[2];
    const float* u   = (const float*)d_in[3];
    float* out = (float*)d_out;

    // one block per batch-head row; 256 threads = 8 wave32
    sortnet_fused_kernel<<<256, 256, 0, stream>>>(q, k, seg, u, out);
}